// CenterPointLidarTarget_83021717832053
// MI455X (gfx1250) — compile-verified
//
#include <hip/hip_runtime.h>
#include <hip/hip_bf16.h>

#define NBATCH 16
#define NOBJ   500
#define NCLS   10
#define FMSZ   360
#define RMAX   12

typedef float v2f __attribute__((ext_vector_type(2)));
typedef float v8f __attribute__((ext_vector_type(8)));

// ---------------- Kernel 1: zero the heatmap with b128 stores ----------------
// 16*10*360*360 = 20,736,000 floats = 5,184,000 float4 = 20,250 blocks * 256.
__global__ void CP_zero_heat(float4* __restrict__ p) {
    size_t i = (size_t)blockIdx.x * blockDim.x + threadIdx.x;
    p[i] = make_float4(0.f, 0.f, 0.f, 0.f);
}

// ---------------- Kernel 2: per-batch stable counting sort by label ----------
__global__ void CP_count_sort(const int* __restrict__ labels, int* __restrict__ order) {
    __shared__ int slab[NOBJ];
    __shared__ int cnt[NCLS];
    __shared__ int start[NCLS];
    const int b = blockIdx.x;
    const int tid = threadIdx.x;
    if (tid < NCLS) cnt[tid] = 0;
    __syncthreads();
    int lab = 0;
    if (tid < NOBJ) {
        lab = labels[b * NOBJ + tid];      // 1-based
        slab[tid] = lab;
        atomicAdd(&cnt[lab - 1], 1);
    }
    __syncthreads();
    if (tid == 0) {
        int s = 0;
        for (int c = 0; c < NCLS; ++c) { start[c] = s; s += cnt[c]; }
    }
    __syncthreads();
    if (tid < NOBJ) {
        int rank = 0;
        for (int j = 0; j < tid; ++j) rank += (slab[j] == lab) ? 1 : 0;
        order[b * NOBJ + start[lab - 1] + rank] = tid;   // stable permutation
    }
}

// ---------------- Kernel 3: one wave32 per sorted object ---------------------
// Gaussian window = outer(v, v): rank-1 update on the WMMA pipe
// (V_WMMA_F32_16X16X4_F32, K=0 column/row carries v, rest zero).
__global__ void CP_splat(const float* __restrict__ boxes,
                         const int* __restrict__ labels,
                         const int* __restrict__ order,
                         float* __restrict__ out) {
    const int lane = threadIdx.x & 31;
    const int wave = (int)((blockIdx.x * blockDim.x + threadIdx.x) >> 5);  // 0..7999
    const int b = wave / NOBJ;
    const int k = wave % NOBJ;

    const int obj = order[b * NOBJ + k];
    const float* bx = boxes + (size_t)(b * NOBJ + obj) * 9;
    const float x = bx[0], y = bx[1], z = bx[2];
    const float w = bx[3], l = bx[4], h = bx[5];
    const float rot = bx[8];
    const int cls = labels[b * NOBJ + obj] - 1;

    const float inv = 1.f / (0.075f * 4.f);             // 1/VOX/OSF
    const float wp = w * inv, lp = l * inv;
    const float cx_f = (x + 54.f) * inv;
    const float cy_f = (y + 54.f) * inv;
    const int cx = (int)cx_f;                           // trunc == .astype(int32)
    const int cy = (int)cy_f;
    const bool valid = (wp > 0.f) && (lp > 0.f) &&
                       (cx >= 0) && (cx < FMSZ) && (cy >= 0) && (cy < FMSZ);

    // gaussian_radius(height=lp, width=wp, 0.1) -- mmdet3d formula verbatim
    const float mo = 0.1f;
    const float hg = lp, wd = wp;
    const float b1 = hg + wd, c1 = wd * hg * (1.f - mo) / (1.f + mo);
    const float r1 = (b1 + sqrtf(b1 * b1 - 4.f * c1)) * 0.5f;
    const float b2 = 2.f * (hg + wd), c2 = (1.f - mo) * wd * hg;
    const float r2 = (b2 + sqrtf(b2 * b2 - 16.f * c2)) * 0.5f;
    const float a3 = 4.f * mo, b3 = -2.f * mo * (hg + wd), c3 = (mo - 1.f) * wd * hg;
    const float r3 = (b3 + sqrtf(b3 * b3 - 4.f * a3 * c3)) * 0.5f;
    const float r = fminf(fminf(r1, r2), r3);
    const int radius = max(2, (int)floorf(r));
    const float sig = (2.f * (float)radius + 1.f) / 6.f;
    const int reff = min(radius, RMAX);                 // window clamp; sigma unclamped

    // ---- output regions (floats) ----
    const size_t HEAT = (size_t)NBATCH * NCLS * FMSZ * FMSZ;   // 20,736,000
    const size_t ANNO = HEAT;
    const size_t INDO = HEAT + (size_t)NBATCH * NOBJ * 8;
    const size_t MSKO = INDO + (size_t)NBATCH * NOBJ;

    if (lane == 0) {
        float4 a0, a1;
        if (valid) {
            a0 = make_float4(cx_f - (float)cx, cy_f - (float)cy, z, logf(w));
            a1 = make_float4(logf(l), logf(h), sinf(rot), cosf(rot));
        } else {
            a0 = make_float4(0.f, 0.f, 0.f, 0.f);
            a1 = a0;
        }
        float4* ap = (float4*)(out + ANNO + (size_t)wave * 8);
        ap[0] = a0;
        ap[1] = a1;
        out[INDO + wave] = valid ? (float)(cy * FMSZ + cx) : 0.f;
        out[MSKO + wave] = valid ? 1.f : 0.f;
    }

    // ---- separable Gaussian: v[i] = exp(-(i-12)^2 / (2 sigma^2)) ----
    const float di = (float)(lane - RMAX);
    const float vL = __expf(-di * di / (2.f * sig * sig));

    float* heat = out + (size_t)(b * NCLS + cls) * FMSZ * FMSZ;

    // 2x2 tiles of 16x16 cover the 25x25 window; WMMA executes unconditionally
    // with full EXEC (ISA requirement); masking happens at the atomic.
#pragma unroll
    for (int ty = 0; ty < 2; ++ty) {
#pragma unroll
        for (int tx = 0; tx < 2; ++tx) {
            const float av = __shfl(vL, (lane & 15) + 16 * ty, 32);
            const float bv = __shfl(vL, (lane & 15) + 16 * tx, 32);
            v2f A;  A[0] = (lane < 16) ? av : 0.f;  A[1] = 0.f;  // A[m][0]=v[m+16ty]
            v2f Bm; Bm[0] = (lane < 16) ? bv : 0.f; Bm[1] = 0.f; // B[0][n]=v[n+16tx]
            v8f C = {};
            C = __builtin_amdgcn_wmma_f32_16x16x4_f32(
                    false, A, false, Bm, (short)0, C, false, false);

            const int nn = lane & 15;
            const int mb = (lane < 16) ? 0 : 8;
#pragma unroll
            for (int i = 0; i < 8; ++i) {
                const int dy = 16 * ty + mb + i - RMAX;
                const int dx = 16 * tx + nn - RMAX;
                const int yy = cy + dy, xx = cx + dx;
                const bool ok = valid &&
                                (dx <= reff) && (-dx <= reff) &&
                                (dy <= reff) && (-dy <= reff) &&
                                (yy >= 0) && (yy < FMSZ) && (xx >= 0) && (xx < FMSZ);
                if (ok) {
                    // non-negative floats: u32 ordering == float ordering
                    atomicMax((unsigned int*)&heat[yy * FMSZ + xx],
                              __float_as_uint(C[i]));
                }
            }
        }
    }
}

extern "C" void kernel_launch(void* const* d_in, const int* in_sizes, int n_in,
                              void* d_out, int out_size, void* d_ws, size_t ws_size,
                              hipStream_t stream) {
    (void)in_sizes; (void)n_in; (void)out_size; (void)ws_size;
    const float* boxes  = (const float*)d_in[0];   // [16,500,9] f32
    const int*   labels = (const int*)d_in[1];     // [16,500] i32 (1-based)
    float* out = (float*)d_out;
    int*   order = (int*)d_ws;                     // [16,500] permutation

    // heatmap = 20,736,000 floats = 5,184,000 float4 -> exactly 20,250 blocks
    CP_zero_heat<<<20250, 256, 0, stream>>>((float4*)out);
    CP_count_sort<<<NBATCH, 512, 0, stream>>>(labels, order);
    // 8000 waves, 8 waves/block -> 1000 blocks
    CP_splat<<<1000, 256, 0, stream>>>(boxes, labels, order, out);
}